// GAT_62483184222887
// MI455X (gfx1250) — compile-verified
//
#include <hip/hip_runtime.h>
#include <hip/hip_bf16.h>
#include <stdint.h>

typedef float v2f __attribute__((ext_vector_type(2)));
typedef float v8f __attribute__((ext_vector_type(8)));

#define HEADS 4
#define NEG_SLOPE 0.2f
#define SM_EPS 1e-16f

// ---------------------------------------------------------------------------
// fp32 WMMA GEMM: C[M,Ncols] = A[M,K] * B[K,Ncols].  M % 16 == 0.
// One wave computes one 16x16 C tile; 4 waves per block -> 64 cols per block.
// K and Ncols are compile-time so all B loads / C stores use immediate offsets.
// A 16x4 f32 layout: lane<16 holds row=lane, K = {0,1}; lane>=16 K = {2,3}.
// B 4x16 f32 layout (symmetric): lane holds col=lane&15, same K split.
// C/D: VGPR i holds (M = i + 8*(lane>>4), N = lane&15).
// ---------------------------------------------------------------------------
template <int K, int NCOLS>
__global__ void k_gemm_wmma(const float* __restrict__ A, const float* __restrict__ B,
                            float* __restrict__ C) {
    const int lane = threadIdx.x & 31;
    const int wave = threadIdx.x >> 5;
    const int row0 = blockIdx.x << 4;
    const int col0 = ((blockIdx.y << 2) + wave) << 4;
    const int r    = lane & 15;
    const int kb   = (lane >> 4) << 1;   // 0 or 2

    v8f acc = {};
    const float* ap = A + (size_t)(row0 + r) * K + kb;
    const float* bp = B + (size_t)kb * NCOLS + col0 + r;
#pragma unroll
    for (int k = 0; k < K; k += 4) {
        v2f a, b;
        a.x = ap[k];
        a.y = ap[k + 1];
        b.x = bp[k * NCOLS];            // compile-time immediate offsets
        b.y = bp[(k + 1) * NCOLS];
        acc = __builtin_amdgcn_wmma_f32_16x16x4_f32(
            /*neg_a=*/false, a, /*neg_b=*/false, b,
            /*c_mod=*/(short)0, acc, /*reuse_a=*/false, /*reuse_b=*/false);
    }
    const int hi = lane >> 4;
    float* cp = C + (size_t)(row0 + (hi << 3)) * NCOLS + col0 + r;
#pragma unroll
    for (int i = 0; i < 8; ++i) cp[i * NCOLS] = acc[i];
}

// ---------------------------------------------------------------------------
// Per-node attention coefficients: asrc[n,h] = <h[n,h,:], att_src[h,:]>.
// ---------------------------------------------------------------------------
__global__ void k_att(const float* __restrict__ Hm, const float* __restrict__ att_s,
                      const float* __restrict__ att_d, float* __restrict__ asrc,
                      float* __restrict__ adst, int n, int ch) {
    int i = blockIdx.x * blockDim.x + threadIdx.x;
    if (i >= n) return;
    const float* hr = Hm + (size_t)i * HEADS * ch;
#pragma unroll
    for (int h = 0; h < HEADS; ++h) {
        float ss = 0.f, sd = 0.f;
        for (int c = 0; c < ch; ++c) {
            float v = hr[h * ch + c];
            ss += v * att_s[h * ch + c];
            sd += v * att_d[h * ch + c];
        }
        asrc[(size_t)i * HEADS + h] = ss;
        adst[(size_t)i * HEADS + h] = sd;
    }
}

// orderable uint encoding of float for atomicMax
__device__ __forceinline__ unsigned f_enc(float f) {
    unsigned u = __float_as_uint(f);
    return (u & 0x80000000u) ? ~u : (u | 0x80000000u);
}
__device__ __forceinline__ float f_dec(unsigned u) {
    unsigned v = (u & 0x80000000u) ? (u & 0x7fffffffu) : ~u;
    return __uint_as_float(v);
}

__global__ void k_zero_u32(unsigned* __restrict__ p, int count) {
    int i = blockIdx.x * blockDim.x + threadIdx.x;
    if (i < count) p[i] = 0u;
}

// out[i] = bias[i % 64]
__global__ void k_bias_fill(float* __restrict__ out, const float* __restrict__ bias,
                            int total) {
    int i = blockIdx.x * blockDim.x + threadIdx.x;
    if (i < total) out[i] = bias[i & 63];
}

__device__ __forceinline__ void edge_sd(const long long* __restrict__ ei, int e, int E,
                                        int& s, int& d) {
    if (e < E) { s = (int)ei[e]; d = (int)ei[(size_t)E + e]; }
    else       { s = e - E; d = e - E; }   // self loop
}

// segment max of leaky_relu(asrc[s]+adst[d]) over dst, encoded atomicMax
__global__ void k_edge_max(const long long* __restrict__ ei, int E, int n,
                           const float* __restrict__ asrc, const float* __restrict__ adst,
                           unsigned* __restrict__ menc) {
    int e = blockIdx.x * blockDim.x + threadIdx.x;
    if (e >= E + n) return;
    int s, d; edge_sd(ei, e, E, s, d);
#pragma unroll
    for (int h = 0; h < HEADS; ++h) {
        float v = asrc[(size_t)s * HEADS + h] + adst[(size_t)d * HEADS + h];
        v = v > 0.f ? v : NEG_SLOPE * v;
        atomicMax(&menc[(size_t)d * HEADS + h], f_enc(v));
    }
}

__global__ void k_decode_max(unsigned* __restrict__ menc, float* __restrict__ m, int count) {
    int i = blockIdx.x * blockDim.x + threadIdx.x;
    if (i < count) m[i] = f_dec(menc[i]);
}

// ex = exp(e - m[dst]); cache per edge; atomicAdd into denom[dst]
__global__ void k_edge_expsum(const long long* __restrict__ ei, int E, int n,
                              const float* __restrict__ asrc, const float* __restrict__ adst,
                              const float* __restrict__ m, float* __restrict__ denom,
                              float* __restrict__ exbuf) {
    int e = blockIdx.x * blockDim.x + threadIdx.x;
    if (e >= E + n) return;
    int s, d; edge_sd(ei, e, E, s, d);
#pragma unroll
    for (int h = 0; h < HEADS; ++h) {
        float v = asrc[(size_t)s * HEADS + h] + adst[(size_t)d * HEADS + h];
        v = v > 0.f ? v : NEG_SLOPE * v;
        float ex = __expf(v - m[(size_t)d * HEADS + h]);
        exbuf[(size_t)e * HEADS + h] = ex;
        atomicAdd(&denom[(size_t)d * HEADS + h], ex);
    }
}

// finalize alpha in place: exbuf[e,h] = scale * exbuf[e,h] / (denom[d,h] + eps)
__global__ void k_alpha(const long long* __restrict__ ei, int E, int n,
                        const float* __restrict__ denom, float* __restrict__ exbuf,
                        float scale) {
    int idx = blockIdx.x * blockDim.x + threadIdx.x;
    if (idx >= (E + n) * HEADS) return;
    int e = idx >> 2;          // HEADS == 4
    int h = idx & 3;
    int s, d; edge_sd(ei, e, E, s, d);
    exbuf[idx] = scale * exbuf[idx] / (denom[(size_t)d * HEADS + h] + SM_EPS);
}

// Layer-1 scatter (concat heads): thread per (edge, channel c in 0..63), head = c>>4.
// out[d*64 + c] += alpha[e, c>>4] * Hsrc[s*64 + c]    -> 1 atomic per thread.
__global__ void k_scatter_concat(const long long* __restrict__ ei, int E, int n,
                                 const float* __restrict__ alpha,
                                 const float* __restrict__ Hsrc, float* __restrict__ out) {
    long long idx = (long long)blockIdx.x * blockDim.x + threadIdx.x;
    if (idx >= (long long)(E + n) * 64) return;
    int e = (int)(idx >> 6);
    int c = (int)(idx & 63);
    int s, d; edge_sd(ei, e, E, s, d);
    float a = alpha[(size_t)e * HEADS + (c >> 4)];
    atomicAdd(&out[(size_t)d * 64 + c], a * Hsrc[(size_t)s * 64 + c]);
}

// Layer-2 scatter (head mean): thread per (edge, channel c in 0..63); the 4 head
// contributions hit the SAME output element, so sum them in registers first.
// out[d*64 + c] += sum_h alpha[e,h] * Hsrc[s*256 + h*64 + c]   -> 1 atomic per thread.
__global__ void k_scatter_mean(const long long* __restrict__ ei, int E, int n,
                               const float4* __restrict__ alpha4,
                               const float* __restrict__ Hsrc, float* __restrict__ out) {
    long long idx = (long long)blockIdx.x * blockDim.x + threadIdx.x;
    if (idx >= (long long)(E + n) * 64) return;
    int e = (int)(idx >> 6);
    int c = (int)(idx & 63);
    int s, d; edge_sd(ei, e, E, s, d);
    float4 al = alpha4[e];
    const float* hp = Hsrc + (size_t)s * 256 + c;
    float v = al.x * hp[0] + al.y * hp[64] + al.z * hp[128] + al.w * hp[192];
    atomicAdd(&out[(size_t)d * 64 + c], v);
}

__global__ void k_relu(float* __restrict__ p, int count) {
    int i = blockIdx.x * blockDim.x + threadIdx.x;
    if (i < count) p[i] = fmaxf(p[i], 0.f);
}

// ---------------------------------------------------------------------------
extern "C" void kernel_launch(void* const* d_in, const int* in_sizes, int n_in,
                              void* d_out, int out_size, void* d_ws, size_t ws_size,
                              hipStream_t stream) {
    const float*     x        = (const float*)d_in[0];
    const long long* ei       = (const long long*)d_in[1];   // int64 [2,E]
    const float*     W1       = (const float*)d_in[2];
    const float*     att_src1 = (const float*)d_in[3];
    const float*     att_dst1 = (const float*)d_in[4];
    const float*     b1       = (const float*)d_in[5];
    const float*     W2       = (const float*)d_in[6];
    const float*     att_src2 = (const float*)d_in[7];
    const float*     att_dst2 = (const float*)d_in[8];
    const float*     b2       = (const float*)d_in[9];
    float* out = (float*)d_out;

    const int IN_CH = 128, HID = 16, OUT_CH = 64;
    const int N = in_sizes[0] / IN_CH;        // 100000
    const int E = in_sizes[1] / 2;            // 1600000
    const int Etot = E + N;

    // workspace layout (floats)
    float* ws    = (float*)d_ws;
    float* h1    = ws;                         // N*64
    float* agg1  = h1   + (size_t)N * 64;      // N*64  (becomes relu'd layer-1 output)
    float* h2    = agg1 + (size_t)N * 64;      // N*256
    float* asrc  = h2   + (size_t)N * 256;     // N*4
    float* adst  = asrc + (size_t)N * HEADS;   // N*4
    float* mbuf  = adst + (size_t)N * HEADS;   // N*4 (unsigned during max pass)
    float* denom = mbuf + (size_t)N * HEADS;   // N*4
    float* exbuf = denom + (size_t)N * HEADS;  // Etot*4 (exp, then alpha in place)

    const int TB = 256;
    dim3 gemmBlk(128);
    int nodeBlocks  = (N + TB - 1) / TB;
    int edgeBlocks  = (Etot + TB - 1) / TB;
    int ehBlocks    = (Etot * HEADS + TB - 1) / TB;
    int nhBlocks    = (N * HEADS + TB - 1) / TB;
    int n64Blocks   = (N * 64 + TB - 1) / TB;
    long long scatThreads = (long long)Etot * 64;
    int scatBlocks  = (int)((scatThreads + TB - 1) / TB);

    // ---------------- Layer 1 ----------------
    k_gemm_wmma<128, 64><<<dim3(N / 16, 1), gemmBlk, 0, stream>>>(x, W1, h1);
    k_att<<<nodeBlocks, TB, 0, stream>>>(h1, att_src1, att_dst1, asrc, adst, N, HID);
    k_zero_u32<<<(N * HEADS * 2 + TB - 1) / TB, TB, 0, stream>>>((unsigned*)mbuf, N * HEADS * 2);
    k_bias_fill<<<n64Blocks, TB, 0, stream>>>(agg1, b1, N * 64);
    k_edge_max<<<edgeBlocks, TB, 0, stream>>>(ei, E, N, asrc, adst, (unsigned*)mbuf);
    k_decode_max<<<nhBlocks, TB, 0, stream>>>((unsigned*)mbuf, mbuf, N * HEADS);
    k_edge_expsum<<<edgeBlocks, TB, 0, stream>>>(ei, E, N, asrc, adst, mbuf, denom, exbuf);
    k_alpha<<<ehBlocks, TB, 0, stream>>>(ei, E, N, denom, exbuf, 1.0f);
    k_scatter_concat<<<scatBlocks, TB, 0, stream>>>(ei, E, N, exbuf, h1, agg1);
    k_relu<<<n64Blocks, TB, 0, stream>>>(agg1, N * 64);

    // ---------------- Layer 2 ----------------
    k_gemm_wmma<64, 256><<<dim3(N / 16, 4), gemmBlk, 0, stream>>>(agg1, W2, h2);
    k_att<<<nodeBlocks, TB, 0, stream>>>(h2, att_src2, att_dst2, asrc, adst, N, OUT_CH);
    k_zero_u32<<<(N * HEADS * 2 + TB - 1) / TB, TB, 0, stream>>>((unsigned*)mbuf, N * HEADS * 2);
    k_bias_fill<<<n64Blocks, TB, 0, stream>>>(out, b2, N * 64);
    k_edge_max<<<edgeBlocks, TB, 0, stream>>>(ei, E, N, asrc, adst, (unsigned*)mbuf);
    k_decode_max<<<nhBlocks, TB, 0, stream>>>((unsigned*)mbuf, mbuf, N * HEADS);
    k_edge_expsum<<<edgeBlocks, TB, 0, stream>>>(ei, E, N, asrc, adst, mbuf, denom, exbuf);
    k_alpha<<<ehBlocks, TB, 0, stream>>>(ei, E, N, denom, exbuf, 0.25f);   // 1/HEADS folded in
    k_scatter_mean<<<scatBlocks, TB, 0, stream>>>(ei, E, N, (const float4*)exbuf, h2, out);
}